// RoIAlignFeatureExtractor_55052890800924
// MI455X (gfx1250) — compile-verified
//
#include <hip/hip_runtime.h>
#include <hip/hip_bf16.h>
#include <math.h>

// Problem constants (fixed by reference setup_inputs)
#define BB   4
#define CC   256
#define HH   256
#define WW   256
#define HWW  (HH * WW)          // 65536
#define PP   7

// Transpose tiling
#define TILE_HW   256
#define TILE_C    32
#define LDS_PITCH 258           // 256 dwords + 2-dword TDM pad per row

typedef unsigned int u32;
typedef unsigned long long u64;
typedef __attribute__((ext_vector_type(4))) unsigned int v4u;
typedef __attribute__((ext_vector_type(8))) int          v8i;
typedef __attribute__((ext_vector_type(4))) int          v4i;

#if defined(__has_builtin)
#if __has_builtin(__builtin_amdgcn_tensor_load_to_lds)
#define HAVE_TDM 1
#endif
#endif

// ---------------------------------------------------------------------------
// Kernel 1: (B,C,H,W) -> (B,H*W,C) transpose. One 32(C) x 256(HW) tile per
// block. Wave 0 drives the Tensor Data Mover; TDM pads LDS by 2 dwords every
// 256 dwords so the column reads below are bank-conflict free.
// ---------------------------------------------------------------------------
__global__ __launch_bounds__(256)
void transpose_tdm(const float* __restrict__ in, float* __restrict__ outp) {
    __shared__ float tile[TILE_C * LDS_PITCH];

    const int hw0 = blockIdx.x * TILE_HW;
    const int c0  = blockIdx.y * TILE_C;
    const int b   = blockIdx.z;

#if HAVE_TDM
    if (threadIdx.x < 32) {   // only wave 0 issues the DMA
        const u32 lds_addr = (u32)(size_t)(void*)tile;  // low 32 bits = LDS byte offset
        const u64 gaddr = (u64)(size_t)(in + ((size_t)(b * CC + c0) * HWW + hw0));

        // ---- D# group 0 (128b): count=1, lds_addr, global_addr, type=2 ----
        v4u g0;
        g0[0] = 1u;                                   // count=1, user mode
        g0[1] = lds_addr;                             // lds_addr [63:32]
        g0[2] = (u32)(gaddr & 0xFFFFFFFFull);         // global_addr[31:0]
        g0[3] = (u32)((gaddr >> 32) & 0x1FFFFFFull)   // global_addr[56:32]
              | (2u << 30);                           // type = 2 ("image")

        // ---- D# group 1 (256b) ----
        // data_size=2 (4B), pad_enable=1, pad_interval=7 (256 dwords),
        // pad_amount=1 (2 dwords) -> LDS row pitch 258 dwords.
        v8i g1;
        g1[0] = (2 << 16) | (1 << 20) | (7 << 22) | (1 << 25);
        g1[1] = (int)(((u32)HWW & 0xFFFFu) << 16);            // tensor_dim0[15:0]  (=0)
        g1[2] = (int)(((u32)HWW >> 16) | ((u32)(BB * CC) << 16)); // dim0 hi | dim1 lo
        g1[3] = (int)(((u32)(BB * CC) >> 16) | ((u32)TILE_HW << 16)); // dim1 hi | tile_dim0
        g1[4] = (int)((u32)TILE_C);                           // tile_dim1 | tile_dim2=0
        g1[5] = (int)((u32)HWW);                              // tensor_dim0_stride lo
        g1[6] = (int)(((u32)HWW & 0xFFFFu) << 16);            // stride0 hi=0 | stride1 lo
        g1[7] = (int)((u32)HWW >> 16);                        // stride1 hi

        // ---- groups 2/3: dims collapsed to 1, tiles 0 (unused) ----
        v4i g2; g2[0] = 1; g2[1] = 1; g2[2] = 0; g2[3] = 0;
        v4i g3; g3[0] = 0; g3[1] = (1 << 16); g3[2] = 0; g3[3] = 0;

        // clang-23 toolchain: 6-arg form with an extra int32x8 operand
        // before the cache-policy immediate; pass it zero-filled.
        v8i g4; g4[0] = 0; g4[1] = 0; g4[2] = 0; g4[3] = 0;
                g4[4] = 0; g4[5] = 0; g4[6] = 0; g4[7] = 0;

        __builtin_amdgcn_tensor_load_to_lds(g0, g1, g2, g3, g4, 0);
        __builtin_amdgcn_s_wait_tensorcnt(0);
    }
#else
    // Fallback: cooperative coalesced fill of the same padded layout.
    for (int c = 0; c < TILE_C; ++c)
        tile[c * LDS_PITCH + threadIdx.x] =
            in[(size_t)(b * CC + c0 + c) * HWW + hw0 + threadIdx.x];
#endif
    __syncthreads();

    // Transposed write-out: 4 consecutive channels per thread -> b128 stores.
    const int cl  = (threadIdx.x & 7) * 4;   // 0..28
    const int hwl = threadIdx.x >> 3;        // 0..31
    float* obase = outp + (size_t)b * HWW * CC;
#pragma unroll
    for (int it = 0; it < 8; ++it) {
        const int hw = hwl + it * 32;
        float4 v;
        v.x = tile[(cl + 0) * LDS_PITCH + hw];
        v.y = tile[(cl + 1) * LDS_PITCH + hw];
        v.z = tile[(cl + 2) * LDS_PITCH + hw];
        v.w = tile[(cl + 3) * LDS_PITCH + hw];
        *(float4*)(obase + (size_t)(hw0 + hw) * CC + c0 + cl) = v;
    }
}

// ---------------------------------------------------------------------------
// Weight/offset computation shared by both gather kernels.
// Exactly mirrors the reference clamping/validity semantics.
// ---------------------------------------------------------------------------
__device__ __forceinline__
void cell_weights(const float* __restrict__ boxes, int roi, int py, int px,
                  int* __restrict__ s_yx, float* __restrict__ s_w) {
    const float gw = 281.6f / (float)WW;   // 1.1
    const float gh = 80.0f  / (float)HH;   // 0.3125
    const float bx = boxes[roi * 7 + 0];
    const float by = boxes[roi * 7 + 1];
    const float l4 = boxes[roi * 7 + 4];
    const float l5 = boxes[roi * 7 + 5];
    const float th = -boxes[roi * 7 + 6];

    const float cx = (bx + 140.8f) / gw - 0.5f;
    const float cy = (by + 40.0f)  / gh - 0.5f;
    const float rw = l5 / gw;
    const float rh = l4 / gh;
    const float binw = rw * (1.0f / PP);
    const float binh = rh * (1.0f / PP);
    float sv, cv;
    sincosf(th, &sv, &cv);

#pragma unroll
    for (int sy = 0; sy < 2; ++sy) {
#pragma unroll
        for (int sx = 0; sx < 2; ++sx) {
            const float yv = -rh * 0.5f + binh * ((float)py + ((float)sy + 0.5f) * 0.5f);
            const float xv = -rw * 0.5f + binw * ((float)px + ((float)sx + 0.5f) * 0.5f);
            const float ys = yv * cv - xv * sv + cy;
            const float xs = yv * sv + xv * cv + cx;
            const bool valid = (ys > -1.0f) && (ys < (float)HH) &&
                               (xs > -1.0f) && (xs < (float)WW);
            const float y = fmaxf(ys, 0.0f);
            const float x = fmaxf(xs, 0.0f);
            int y0 = min((int)floorf(y), HH - 1);
            int x0 = min((int)floorf(x), WW - 1);
            const int y1 = min(y0 + 1, HH - 1);
            const int x1 = min(x0 + 1, WW - 1);
            const float yf = (y0 >= HH - 1) ? (float)y0 : y;
            const float xf = (x0 >= WW - 1) ? (float)x0 : x;
            const float ly = yf - (float)y0, lx = xf - (float)x0;
            const float hy = 1.0f - ly,      hx = 1.0f - lx;
            const float s = valid ? 0.25f : 0.0f;   // fold S*S mean + validity
            const int q = (sy * 2 + sx) * 4;
            s_yx[q + 0] = y0 * WW + x0;
            s_yx[q + 1] = y0 * WW + x1;
            s_yx[q + 2] = y1 * WW + x0;
            s_yx[q + 3] = y1 * WW + x1;
            s_w[q + 0] = hy * hx * s;
            s_w[q + 1] = hy * lx * s;
            s_w[q + 2] = ly * hx * s;
            s_w[q + 3] = ly * lx * s;
        }
    }
}

// ---------------------------------------------------------------------------
// Kernel 2: gather from channel-last fmapT. One block per output cell,
// 256 threads = 256 channels -> all 16 corner fetches are coalesced 1KB bursts.
// ---------------------------------------------------------------------------
__global__ __launch_bounds__(256)
void roi_gather(const float* __restrict__ fmapT, const float* __restrict__ boxes,
                float* __restrict__ outp, int M) {
    __shared__ int   s_yx[16];
    __shared__ float s_w[16];

    const int cell = blockIdx.x;
    const int px  = cell % PP;
    const int py  = (cell / PP) % PP;
    const int roi = cell / (PP * PP);
    const int b   = roi / M;

    if (threadIdx.x == 0) cell_weights(boxes, roi, py, px, s_yx, s_w);
    __syncthreads();

    const int c = threadIdx.x;
    const size_t base = (size_t)b * HWW;
    float acc = 0.0f;
#pragma unroll
    for (int k = 0; k < 16; ++k)
        acc = fmaf(s_w[k], fmapT[((base + (size_t)s_yx[k]) << 8) + c], acc);

    outp[(size_t)cell * CC + c] = acc;
}

// ---------------------------------------------------------------------------
// Fallback: gather straight from (B,C,H,W) if workspace is too small.
// Correct but strided; only used when ws_size < 256 MB.
// ---------------------------------------------------------------------------
__global__ __launch_bounds__(256)
void roi_gather_direct(const float* __restrict__ feat, const float* __restrict__ boxes,
                       float* __restrict__ outp, int M) {
    __shared__ int   s_yx[16];
    __shared__ float s_w[16];

    const int cell = blockIdx.x;
    const int px  = cell % PP;
    const int py  = (cell / PP) % PP;
    const int roi = cell / (PP * PP);
    const int b   = roi / M;

    if (threadIdx.x == 0) cell_weights(boxes, roi, py, px, s_yx, s_w);
    __syncthreads();

    const int c = threadIdx.x;
    const size_t chbase = (size_t)(b * CC + c) * HWW;
    float acc = 0.0f;
#pragma unroll
    for (int k = 0; k < 16; ++k)
        acc = fmaf(s_w[k], feat[chbase + (size_t)s_yx[k]], acc);

    outp[(size_t)cell * CC + c] = acc;
}

// ---------------------------------------------------------------------------
extern "C" void kernel_launch(void* const* d_in, const int* in_sizes, int n_in,
                              void* d_out, int out_size, void* d_ws, size_t ws_size,
                              hipStream_t stream) {
    const float* feature = (const float*)d_in[0];
    const float* boxes   = (const float*)d_in[1];
    float* out = (float*)d_out;

    const int M = in_sizes[1] / (BB * 7);           // 100
    const int cells = BB * M * PP * PP;             // 19600
    const size_t need = (size_t)BB * HWW * CC * sizeof(float);  // 256 MB

    if (ws_size >= need) {
        float* fmapT = (float*)d_ws;
        dim3 gt(HWW / TILE_HW, CC / TILE_C, BB);    // (256, 8, 4)
        transpose_tdm<<<gt, 256, 0, stream>>>(feature, fmapT);
        roi_gather<<<cells, 256, 0, stream>>>(fmapT, boxes, out, M);
    } else {
        roi_gather_direct<<<cells, 256, 0, stream>>>(feature, boxes, out, M);
    }
}